// LFSa_15384572854576
// MI455X (gfx1250) — compile-verified
//
#include <hip/hip_runtime.h>

typedef __attribute__((ext_vector_type(2))) float v2f;
typedef __attribute__((ext_vector_type(8))) float v8f;

#define ATT_STRIDE 132   // 128 + 4 pad (row-major X / P region)
#define ATT_STRIDE2 264  // interleaved row-pair stride (= 2*ATT_STRIDE)
#define XS_STRIDE2 136   // conv1x1 interleaved row-pair stride (2*68)

static __device__ __forceinline__ v8f wmma_f32(v2f a, v2f b, v8f c) {
  // V_WMMA_F32_16X16X4_F32: D = A(16x4) * B(4x16) + C(16x16), fp32 throughout
  return __builtin_amdgcn_wmma_f32_16x16x4_f32(false, a, false, b, (short)0, c,
                                               false, false);
}

// ---------------------------------------------------------------------------
// 1x1 conv as channel GEMM: out[b,o,p] = sum_c W[o,c]*in[b,c,p] + bias[o]
// Block: 256 threads (8 waves), one batch, 64 spatial positions.
// Staging buffer is stored row-pair interleaved so every B fragment
// (in[k][n], in[k+1][n]) is one aligned ds_load_b64.
// ---------------------------------------------------------------------------
template <bool CONCAT>
__global__ __launch_bounds__(256) void conv1_kernel(
    const float* __restrict__ in0, const float* __restrict__ in1,
    const float* __restrict__ wmat, const float* __restrict__ bias,
    float* __restrict__ out) {
  __shared__ float xs[64 * XS_STRIDE2];  // addr(c,n) = (c>>1)*136 + 2n + (c&1)
  const int b  = blockIdx.y;
  const int n0 = blockIdx.x * 64;
  const int t  = threadIdx.x;

#pragma unroll
  for (int i = 0; i < 32; ++i) {
    int idx = t + i * 256;          // 0..8191
    int c = idx >> 6, n = idx & 63;
    float v;
    if (CONCAT) {
      v = (c < 64) ? in0[(((size_t)b * 64 + c) << 14) + n0 + n]
                   : in1[(((size_t)b * 64 + (c - 64)) << 14) + n0 + n];
    } else {
      v = in0[(((size_t)b * 128 + c) << 14) + n0 + n];
    }
    xs[(c >> 1) * XS_STRIDE2 + 2 * n + (c & 1)] = v;
  }
  __syncthreads();

  const int wave = t >> 5, lane = t & 31;
  const int o0 = wave * 16;
  const int laneM = lane & 15, hi = lane >> 4;

  v8f vz = {0.f, 0.f, 0.f, 0.f, 0.f, 0.f, 0.f, 0.f};
  v8f acc[4] = {vz, vz, vz, vz};

  for (int k = 0; k < 128; k += 4) {
    // A fragment: W[o0+laneM, k + 2*hi + {0,1}] — aligned global_load_b64
    v2f a = *(const v2f*)(wmat + (o0 + laneM) * 128 + k + 2 * hi);
    const float* brow = &xs[((k + 2 * hi) >> 1) * XS_STRIDE2];
#pragma unroll
    for (int tn = 0; tn < 4; ++tn) {
      int n = tn * 16 + laneM;
      v2f bb = *(const v2f*)(brow + 2 * n);  // (in[k'][n], in[k'+1][n])
      acc[tn] = wmma_f32(a, bb, acc[tn]);
    }
  }

  // C/D layout: VGPR v, lane: row = v + 8*hi, col = laneM
#pragma unroll
  for (int tn = 0; tn < 4; ++tn) {
#pragma unroll
    for (int v = 0; v < 8; ++v) {
      int o = o0 + v + 8 * hi;
      out[(((size_t)b * 128 + o) << 14) + n0 + tn * 16 + laneM] =
          acc[tn][v] + bias[o];
    }
  }
}

// ---------------------------------------------------------------------------
// Per-(b,c) axial self-attention on a 128x128 tile (q=k=v=X):
//   S = X*X^T * 128^-0.5 ; A = softmax_rows(S) ; O = A*X
// TRANSPOSE=true operates on X^T (column attention), output kept in [W,H].
// One block per (b,c). 8 waves; each wave owns 16 rows.
// LDS: X (row-major, reused as P after the S-pass) + Xil (row-pair
// interleaved copy for the O-pass B fragments). All fragment loads are
// aligned 8-byte ds_load_b64.
// ---------------------------------------------------------------------------
template <bool TRANSPOSE>
__global__ __launch_bounds__(256) void att_kernel(
    const float* __restrict__ qkv, float* __restrict__ outp) {
  __shared__ float X[128 * ATT_STRIDE];    // X for S-pass; P for O-pass
  __shared__ float Xil[64 * ATT_STRIDE2];  // addr(r,c) = (r>>1)*264+2c+(r&1)

  const int bc = blockIdx.x;               // (b*C + c)
  const float* Xg = qkv + (size_t)bc * 16384;
  const int t = threadIdx.x;

#pragma unroll
  for (int i = 0; i < 64; ++i) {
    int idx = t + i * 256;
    int r, c;
    if (TRANSPOSE) { r = idx & 127; c = idx >> 7; }   // coalesced transpose read
    else           { r = idx >> 7;  c = idx & 127; }
    float v = TRANSPOSE ? Xg[c * 128 + r] : Xg[r * 128 + c];
    X[r * ATT_STRIDE + c] = v;
    Xil[(r >> 1) * ATT_STRIDE2 + 2 * c + (r & 1)] = v;
  }
  __syncthreads();

  const int wave = t >> 5, lane = t & 31;
  const int laneM = lane & 15, hi = lane >> 4;
  const int m0 = wave * 16;

  v8f vz = {0.f, 0.f, 0.f, 0.f, 0.f, 0.f, 0.f, 0.f};
  v8f acc[8];
#pragma unroll
  for (int i = 0; i < 8; ++i) acc[i] = vz;

  // ---- S = X * X^T : wave's 16 rows x all 128 cols, K = 128 ----
  for (int k = 0; k < 128; k += 4) {
    v2f a = *(const v2f*)&X[(m0 + laneM) * ATT_STRIDE + k + 2 * hi];
#pragma unroll
    for (int tn = 0; tn < 8; ++tn) {
      // B[kk][n] = X[n][kk] — contiguous along kk
      v2f b = *(const v2f*)&X[(tn * 16 + laneM) * ATT_STRIDE + k + 2 * hi];
      acc[tn] = wmma_f32(a, b, acc[tn]);
    }
  }

  // ---- softmax over each row, entirely in registers (scale = 128^-0.5) ----
  const float scale = 0.08838834764831845f;
#pragma unroll
  for (int v = 0; v < 8; ++v) {
    // row = m0 + v + 8*hi ; its 128 cols live in acc[0..7][v] across 16 lanes
    float mx = acc[0][v];
#pragma unroll
    for (int tn = 1; tn < 8; ++tn) mx = fmaxf(mx, acc[tn][v]);
#pragma unroll
    for (int off = 8; off >= 1; off >>= 1)
      mx = fmaxf(mx, __shfl_xor(mx, off, 32));   // stays within 16-lane half
    float sum = 0.f;
#pragma unroll
    for (int tn = 0; tn < 8; ++tn) {
      float e = __expf((acc[tn][v] - mx) * scale);
      acc[tn][v] = e;
      sum += e;
    }
#pragma unroll
    for (int off = 8; off >= 1; off >>= 1) sum += __shfl_xor(sum, off, 32);
    float inv = 1.0f / sum;
#pragma unroll
    for (int tn = 0; tn < 8; ++tn) acc[tn][v] *= inv;
  }

  // Retire all S-pass reads of X, then overwrite X region with P
  // (each wave writes only its own 16 rows; O-pass reads only its own slab).
  __syncthreads();
#pragma unroll
  for (int v = 0; v < 8; ++v) {
    float* Pl = &X[(m0 + v + 8 * hi) * ATT_STRIDE];
#pragma unroll
    for (int tn = 0; tn < 8; ++tn) Pl[tn * 16 + laneM] = acc[tn][v];
  }

  // ---- O = P * X : wave's 16 rows, K = 128 ----
#pragma unroll
  for (int i = 0; i < 8; ++i) acc[i] = vz;
  for (int k = 0; k < 128; k += 4) {
    v2f a = *(const v2f*)&X[(m0 + laneM) * ATT_STRIDE + k + 2 * hi];  // P rows
    const float* brow = &Xil[((k + 2 * hi) >> 1) * ATT_STRIDE2];
#pragma unroll
    for (int tn = 0; tn < 8; ++tn) {
      int n = tn * 16 + laneM;
      v2f b = *(const v2f*)(brow + 2 * n);  // (X[k'][n], X[k'+1][n])
      acc[tn] = wmma_f32(a, b, acc[tn]);
    }
  }

  float* Og = outp + (size_t)bc * 16384;
#pragma unroll
  for (int tn = 0; tn < 8; ++tn) {
#pragma unroll
    for (int v = 0; v < 8; ++v) {
      int row = m0 + v + 8 * hi;
      Og[row * 128 + tn * 16 + laneM] = acc[tn][v];
    }
  }
}

// ---------------------------------------------------------------------------
// Depthwise 7x7 conv (zero pad 3) + residual: out = concat(x1,x2) + dw(y) + lb
// ---------------------------------------------------------------------------
__global__ __launch_bounds__(256) void dw_add_kernel(
    const float* __restrict__ y, const float* __restrict__ x1,
    const float* __restrict__ x2, const float* __restrict__ lw,
    const float* __restrict__ lb, float* __restrict__ out) {
  __shared__ float tile[22 * 24];
  __shared__ float wk[49];
  const int c = blockIdx.y, b = blockIdx.z;
  const int th = blockIdx.x >> 3, tw = blockIdx.x & 7;
  const int h0 = th * 16, w0 = tw * 16;
  const int t = threadIdx.x;
  if (t < 49) wk[t] = lw[c * 49 + t];
  const float* img = y + (((size_t)b * 128 + c) << 14);
  for (int idx = t; idx < 22 * 22; idx += 256) {
    int r = idx / 22, cc = idx % 22;
    int gh = h0 - 3 + r, gw = w0 - 3 + cc;
    float v = 0.f;
    if (gh >= 0 && gh < 128 && gw >= 0 && gw < 128) v = img[gh * 128 + gw];
    tile[r * 24 + cc] = v;
  }
  __syncthreads();
  const int ty = t >> 4, tx = t & 15;
  float acc = lb[c];
#pragma unroll
  for (int i = 0; i < 7; ++i)
#pragma unroll
    for (int j = 0; j < 7; ++j)
      acc = fmaf(tile[(ty + i) * 24 + tx + j], wk[i * 7 + j], acc);
  const int h = h0 + ty, w = w0 + tx;
  float xv = (c < 64) ? x1[(((size_t)b * 64 + c) << 14) + h * 128 + w]
                      : x2[(((size_t)b * 64 + (c - 64)) << 14) + h * 128 + w];
  out[(((size_t)b * 128 + c) << 14) + h * 128 + w] = xv + acc;
}

// ---------------------------------------------------------------------------
// Depthwise 7x7 on the [W,H]-layout column branch, accumulated transposed:
//   out[b,c,h,w] += (dw(y_col))[b,c,w,h] + lb[c]
// LDS transpose keeps global stores coalesced.
// ---------------------------------------------------------------------------
__global__ __launch_bounds__(256) void dw_addT_kernel(
    const float* __restrict__ y, const float* __restrict__ lw,
    const float* __restrict__ lb, float* __restrict__ out) {
  __shared__ float tile[22 * 24];
  __shared__ float wk[49];
  __shared__ float zt[16 * 17];
  const int c = blockIdx.y, b = blockIdx.z;
  const int tr = blockIdx.x >> 3, tc = blockIdx.x & 7;
  const int r0 = tr * 16, c0 = tc * 16;  // r = w-dim, c = h-dim
  const int t = threadIdx.x;
  if (t < 49) wk[t] = lw[c * 49 + t];
  const float* img = y + (((size_t)b * 128 + c) << 14);  // [W,H] image
  for (int idx = t; idx < 22 * 22; idx += 256) {
    int r = idx / 22, cc = idx % 22;
    int gr = r0 - 3 + r, gc = c0 - 3 + cc;
    float v = 0.f;
    if (gr >= 0 && gr < 128 && gc >= 0 && gc < 128) v = img[gr * 128 + gc];
    tile[r * 24 + cc] = v;
  }
  __syncthreads();
  const int ty = t >> 4, tx = t & 15;
  float acc = lb[c];
#pragma unroll
  for (int i = 0; i < 7; ++i)
#pragma unroll
    for (int j = 0; j < 7; ++j)
      acc = fmaf(tile[(ty + i) * 24 + tx + j], wk[i * 7 + j], acc);
  zt[ty * 17 + tx] = acc;  // z at (w = r0+ty, h = c0+tx)
  __syncthreads();
  // thread writes (h = c0+ty, w = r0+tx) -> coalesced over tx
  const size_t oidx =
      (((size_t)b * 128 + c) << 14) + (size_t)(c0 + ty) * 128 + r0 + tx;
  out[oidx] += zt[tx * 17 + ty];
}

// ---------------------------------------------------------------------------
extern "C" void kernel_launch(void* const* d_in, const int* in_sizes, int n_in,
                              void* d_out, int out_size, void* d_ws,
                              size_t ws_size, hipStream_t stream) {
  (void)in_sizes; (void)n_in; (void)out_size; (void)ws_size;
  const float* x1   = (const float*)d_in[0];
  const float* x2   = (const float*)d_in[1];
  const float* c_w  = (const float*)d_in[2];
  const float* c_b  = (const float*)d_in[3];
  const float* sc_w = (const float*)d_in[4];
  const float* sc_b = (const float*)d_in[5];
  const float* lc_w = (const float*)d_in[6];
  const float* lc_b = (const float*)d_in[7];
  float* out = (float*)d_out;

  const size_t N = (size_t)16 * 128 * 128 * 128;  // 33,554,432 elements
  float* buf0 = (float*)d_ws;       // qkv, later sc(x_row)
  float* buf1 = buf0 + N;           // x_row, later sc(x_col)
  float* buf2 = buf1 + N;           // x_col in [W,H]

  dim3 cgrid(256, 16);              // 64 spatial cols per block, 16 batches
  conv1_kernel<true><<<cgrid, 256, 0, stream>>>(x1, x2, c_w, c_b, buf0);

  att_kernel<false><<<2048, 256, 0, stream>>>(buf0, buf1);  // row attention
  att_kernel<true><<<2048, 256, 0, stream>>>(buf0, buf2);   // col attention

  conv1_kernel<false><<<cgrid, 256, 0, stream>>>(buf1, nullptr, sc_w, sc_b, buf0);
  conv1_kernel<false><<<cgrid, 256, 0, stream>>>(buf2, nullptr, sc_w, sc_b, buf1);

  dim3 dgrid(64, 128, 16);          // 8x8 tiles, C channels, B batches
  dw_add_kernel<<<dgrid, 256, 0, stream>>>(buf0, x1, x2, lc_w, lc_b, out);
  dw_addT_kernel<<<dgrid, 256, 0, stream>>>(buf1, lc_w, lc_b, out);
}